// NCFG_21139829031662
// MI455X (gfx1250) — compile-verified
//
#include <hip/hip_runtime.h>
#include <hip/hip_bf16.h>

#define DIM   32
#define KTRIP 64
#define LHOPS 2

typedef float v2f __attribute__((ext_vector_type(2)));
typedef float v8f __attribute__((ext_vector_type(8)));

static __device__ __forceinline__ v8f wmma4(v2f a, v2f b, v8f c) {
  // D = A(16x4 f32) * B(4x16 f32) + C(16x16 f32)
  return __builtin_amdgcn_wmma_f32_16x16x4_f32(
      /*neg_a=*/false, a, /*neg_b=*/false, b,
      /*c_mod=*/(short)0, c, /*reuse_a=*/false, /*reuse_b=*/false);
}

static __device__ __forceinline__ float fast_tanh(float x) {
#if __has_builtin(__builtin_amdgcn_tanhf)
  return __builtin_amdgcn_tanhf(x);          // v_tanh_f32 (gfx1250 TRANS op)
#else
  float e = __expf(2.f * x);                 // native v_exp_f32 path
  return 1.f - 2.f / (e + 1.f);
#endif
}

__global__ __launch_bounds__(128)
void ripple_kernel(const int* __restrict__ users,
                   const int* __restrict__ items,
                   const int* __restrict__ hop0,
                   const int* __restrict__ heads,
                   const int* __restrict__ rels,
                   const int* __restrict__ tails,
                   const float* __restrict__ entity_emb,
                   const float* __restrict__ relation_emb,
                   const float* __restrict__ rec_user_emb,
                   const float* __restrict__ rec_item_emb,
                   const float* __restrict__ W_ih,   // [32][64] row-major
                   const float* __restrict__ W_hh,   // [32][32] row-major (indexed [e][d])
                   const float* __restrict__ b_ih,
                   const float* __restrict__ b_hh,
                   float* __restrict__ out,
                   int Btot)
{
  __shared__ __align__(16) float sWih[32 * 64];
  __shared__ __align__(16) float sWhh[32 * 32];
  __shared__ __align__(16) float sH1[4 * 16 * 32];   // per-wave 16x32 staging
  __shared__ __align__(16) float sLog[KTRIP];        // logits, then normalized pi
  __shared__ __align__(16) float sO[DIM];

  const int b    = blockIdx.x;
  const int tid  = threadIdx.x;
  const int w    = tid >> 5;        // wave id 0..3
  const int lane = tid & 31;
  const int mrow = lane & 15;       // row within 16-row tile
  const int hi   = lane >> 4;       // half-wave: K-offset 2*hi in WMMA fragments

  // ---- stage weights in LDS, zero accumulator ----
  for (int i = tid; i < 32 * 64; i += 128) sWih[i] = W_ih[i];
  for (int i = tid; i < 32 * 32; i += 128) sWhh[i] = W_hh[i];
  if (tid < DIM) sO[tid] = 0.f;
  __syncthreads();

  // ---- hop-0: o = sum_k entity_emb[hop0[b,k]] ----
  {
    const int d = tid & 31;
    const int g = tid >> 5;
    float acc = 0.f;
    for (int k = g; k < KTRIP; k += 4)
      acc += entity_emb[(size_t)hop0[(size_t)b * KTRIP + k] * DIM + d];
    atomicAdd(&sO[d], acc);
  }

  // per-lane bias for output dims d = 16n + mrow
  const float bias0 = b_ih[mrow] + b_hh[mrow];
  const float bias1 = b_ih[16 + mrow] + b_hh[16 + mrow];

  const v2f* wihB = (const v2f*)sWih;
  const v2f* whhB = (const v2f*)sWhh;
  float* myH1 = sH1 + w * (16 * 32);

  for (int l = 0; l < LHOPS; ++l) {
    const int row = w * 16 + mrow;                        // triple index k within user
    const size_t base = (size_t)l * Btot * KTRIP + (size_t)b * KTRIP + row;
    const int hidx = heads[base];
    const int ridx = rels[base];
    const int tidx = tails[base];

    // ---- gather hr/tr rows directly into WMMA f32 A-layout (float2 per K-chunk) ----
    v2f hrA[16], trA[16];
    {
      const v2f* hp = (const v2f*)(entity_emb   + (size_t)hidx * DIM);
      const v2f* tp = (const v2f*)(entity_emb   + (size_t)tidx * DIM);
      const v2f* rp = (const v2f*)(relation_emb + (size_t)ridx * DIM);
#pragma unroll
      for (int c = 0; c < 8; ++c) { hrA[c] = hp[2 * c + hi]; trA[c] = tp[2 * c + hi]; }
#pragma unroll
      for (int c = 0; c < 8; ++c) { v2f r = rp[2 * c + hi]; hrA[8 + c] = r; trA[8 + c] = r; }
    }

    // ---- attention logit: (hr . tr) per row ----
    float dotp = 0.f;
#pragma unroll
    for (int c = 0; c < 16; ++c) dotp += hrA[c].x * trA[c].x + hrA[c].y * trA[c].y;
    dotp += __shfl_xor(dotp, 16, 32);
    if (hi == 0) sLog[row] = dotp;

    // ---- GEMM1: h1 = tanh(hr * W_ih^T + b) via f32 WMMA ----
    v8f h1acc[2];
#pragma unroll
    for (int n = 0; n < 2; ++n) {
      v8f acc = {};
      const int bbase = (16 * n + mrow) * 32 + hi;   // float2 index into sWih row
#pragma unroll
      for (int c = 0; c < 16; ++c)
        acc = wmma4(hrA[c], wihB[bbase + 2 * c], acc);
      const float bs = n ? bias1 : bias0;
#pragma unroll
      for (int j = 0; j < 8; ++j) acc[j] = fast_tanh(acc[j] + bs);
      h1acc[n] = acc;
    }

    // ---- D-layout -> LDS row-major so we can reload h1 as an A fragment ----
#pragma unroll
    for (int n = 0; n < 2; ++n)
#pragma unroll
      for (int j = 0; j < 8; ++j)
        myH1[(j + 8 * hi) * 32 + 16 * n + mrow] = h1acc[n][j];
    __syncthreads();   // covers sLog + sH1

    // ---- cooperative softmax: wave 0 turns sLog into normalized pi in place ----
    if (w == 0) {
      float a = sLog[lane];
      float c = sLog[lane + 32];
      float m = fmaxf(a, c);
      m = fmaxf(m, __shfl_xor(m, 16, 32));
      m = fmaxf(m, __shfl_xor(m, 8, 32));
      m = fmaxf(m, __shfl_xor(m, 4, 32));
      m = fmaxf(m, __shfl_xor(m, 2, 32));
      m = fmaxf(m, __shfl_xor(m, 1, 32));
      float e0 = __expf(a - m);
      float e1 = __expf(c - m);
      float s = e0 + e1;
      s += __shfl_xor(s, 16, 32);
      s += __shfl_xor(s, 8, 32);
      s += __shfl_xor(s, 4, 32);
      s += __shfl_xor(s, 2, 32);
      s += __shfl_xor(s, 1, 32);
      const float inv = 1.f / s;
      sLog[lane]      = e0 * inv;
      sLog[lane + 32] = e1 * inv;
    }

    // ---- reload h1 as A fragment ----
    v2f h1A[8];
    {
      const v2f* h1p = (const v2f*)myH1;
#pragma unroll
      for (int c = 0; c < 8; ++c) h1A[c] = h1p[mrow * 16 + 2 * c + hi];
    }

    // ---- GEMM2: h2 = tanh(tr * W_ih^T + h1 * W_hh^T + b) ----
    v8f h2acc[2];
#pragma unroll
    for (int n = 0; n < 2; ++n) {
      v8f acc = {};
      const int bbase  = (16 * n + mrow) * 32 + hi;
      const int bbase2 = (16 * n + mrow) * 16 + hi;
#pragma unroll
      for (int c = 0; c < 16; ++c)
        acc = wmma4(trA[c], wihB[bbase + 2 * c], acc);
#pragma unroll
      for (int c = 0; c < 8; ++c)
        acc = wmma4(h1A[c], whhB[bbase2 + 2 * c], acc);
      const float bs = n ? bias1 : bias0;
#pragma unroll
      for (int j = 0; j < 8; ++j) acc[j] = fast_tanh(acc[j] + bs);
      h2acc[n] = acc;
    }
    __syncthreads();   // pi ready in sLog; everyone past GEMM2

    // ---- o += sum_k pi_k * h2_k  (D layout: vgpr j -> row j + 8*hi, col 16n+mrow) ----
#pragma unroll
    for (int n = 0; n < 2; ++n) {
      float s = 0.f;
#pragma unroll
      for (int j = 0; j < 8; ++j) {
        const int kk = w * 16 + j + 8 * hi;
        s += sLog[kk] * h2acc[n][j];
      }
      atomicAdd(&sO[16 * n + mrow], s);
    }
    __syncthreads();   // sLog/sH1 reused next hop; sO atomics settle
  }

  // ---- final score: sigmoid((o + user_emb) . (entity+item emb)) ----
  if (tid < 32) {
    const int d  = tid;
    const int u  = users[b];
    const int it = items[b];
    float uv = sO[d] + rec_user_emb[(size_t)u * DIM + d];
    float iv = entity_emb[(size_t)it * DIM + d] + rec_item_emb[(size_t)it * DIM + d];
    float p = uv * iv;
    p += __shfl_xor(p, 16, 32);
    p += __shfl_xor(p, 8, 32);
    p += __shfl_xor(p, 4, 32);
    p += __shfl_xor(p, 2, 32);
    p += __shfl_xor(p, 1, 32);
    if (d == 0) out[b] = 1.f / (1.f + __expf(-p));
  }
}

extern "C" void kernel_launch(void* const* d_in, const int* in_sizes, int n_in,
                              void* d_out, int out_size, void* d_ws, size_t ws_size,
                              hipStream_t stream) {
  const int*   users        = (const int*)d_in[0];
  const int*   items        = (const int*)d_in[1];
  const int*   hop0         = (const int*)d_in[2];
  const int*   heads        = (const int*)d_in[3];
  const int*   rels         = (const int*)d_in[4];
  const int*   tails        = (const int*)d_in[5];
  const float* entity_emb   = (const float*)d_in[6];
  const float* relation_emb = (const float*)d_in[7];
  const float* rec_user_emb = (const float*)d_in[8];
  const float* rec_item_emb = (const float*)d_in[9];
  const float* W_ih         = (const float*)d_in[10];
  const float* W_hh         = (const float*)d_in[11];
  const float* b_ih         = (const float*)d_in[12];
  const float* b_hh         = (const float*)d_in[13];
  float* out = (float*)d_out;

  const int B = in_sizes[0];   // 4096
  ripple_kernel<<<B, 128, 0, stream>>>(users, items, hop0, heads, rels, tails,
                                       entity_emb, relation_emb,
                                       rec_user_emb, rec_item_emb,
                                       W_ih, W_hh, b_ih, b_hh, out, B);
}